// ReversibleNetwork_71339406787230
// MI455X (gfx1250) — compile-verified
//
#include <hip/hip_runtime.h>
#include <stdint.h>

#define NBATCH     4096
#define NIN        784
#define NOUT       784
#define FWIDTH     1024
#define FHALF      512
#define NDEPTH     8
#define NBDEPTH    10
#define SDIM       20
#define ROWS       8
#define NTHREADS   256
#define F4_PER_ROW (NIN/4)      /* 196 */
#define PADCOLS    (FWIDTH-NIN) /* 240 */
#define CS_TAB_ELEMS (NDEPTH*NBDEPTH*FHALF)          /* 40960 float2 */
#define CS_TAB_BYTES (CS_TAB_ELEMS*2*sizeof(float))  /* 327680 B */

__constant__ float c_T[23] = {
  -8.f,-8.f,-8.f,-4.f,-2.f,-1.f,-0.5f,-0.25f,-0.125f,-0.0625f,-0.03125f,
  0.f,0.03125f,0.0625f,0.125f,0.25f,0.5f,1.f,2.f,4.f,8.f,8.f,8.f
};

// CDNA5 async copy: global memory -> LDS, tracked by ASYNCcnt.
__device__ __forceinline__ void async_ld_b128(unsigned lds_off, const void* g) {
  asm volatile("global_load_async_to_lds_b128 %0, %1, off"
               :: "v"(lds_off), "v"(g) : "memory");
}
__device__ __forceinline__ void wait_async0() {
  asm volatile("s_wait_asynccnt 0" ::: "memory");
}

// One-shot: cos/sin of every rotation angle -> d_ws, layout [layer][d][k].
__global__ __launch_bounds__(NTHREADS)
void sincos_tab_kernel(const float* __restrict__ bp, float2* __restrict__ tab) {
  int t = blockIdx.x * NTHREADS + threadIdx.x;
  if (t >= NDEPTH * FHALF * NBDEPTH) return;
  int i = t / (FHALF * NBDEPTH);
  int r = t - i * (FHALF * NBDEPTH);
  int k = r / NBDEPTH;
  int d = r - k * NBDEPTH;
  float sv, cv;
  sincosf(bp[t], &sv, &cv);                  // bp[t] == theta(layer i, col k, stage d)
  tab[((size_t)i * NBDEPTH + d) * FHALF + k] = make_float2(cv, sv);
}

// Interval index + 3 non-zero degree-2 basis values via exponent bits + rcp table.
__device__ __forceinline__ float spline_eval(float xv, const float* __restrict__ cf,
                                             const float (*jt)[8]) {
  float x = fminf(fmaxf(xv, -8.0f), 7.9999990463256836f);
  unsigned u = __float_as_uint(x);
  float ax = __uint_as_float(u & 0x7fffffffu);
  int  e    = (int)((u >> 23) & 0xffu) - 127;      // floor(log2 |x|)
  bool man0 = (u & 0x7fffffu) == 0u;               // |x| exact power of two
  int jpos = (ax < 0.03125f) ? 11 : (e + 17);
  int en   = man0 ? (e - 1) : e;
  int jneg = (ax <= 0.03125f) ? 10 : (4 - en);
  int j    = (x < 0.0f) ? jneg : jpos;             // -0.0 -> jpos (x>=0 semantics)
  float4 a = *(const float4*)&jt[j][0];            // t[j-1], t[j], t[j+1], t[j+2]
  float4 b = *(const float4*)&jt[j][4];            // 1/d10, 1/d20, 1/d21, 0
  float B1j   = (x - a.y) * b.x;
  float B1jm1 = (a.z - x) * b.x;
  float B2j   = ((x - a.y) * b.z) * B1j;
  float B2jm1 = ((x - a.x) * b.y) * B1jm1 + ((a.w - x) * b.z) * B1j;
  float B2jm2 = ((a.z - x) * b.y) * B1jm1;
  return B2jm2*cf[j-2] + B2jm1*cf[j-1] + B2j*cf[j];
}

template<bool USE_TAB>
__global__ __launch_bounds__(NTHREADS)
void revnet_kernel(const float* __restrict__ X, const float* __restrict__ bp,
                   const float2* __restrict__ tab, const float* __restrict__ sc,
                   float* __restrict__ out)
{
  __shared__ __align__(16) float bufA[ROWS*FWIDTH];
  __shared__ __align__(16) float bufB[ROWS*FWIDTH];
  __shared__ __align__(16) float coefLDS[FHALF*SDIM];
  __shared__ __align__(16) float jtab[20][8];

  const int tid = threadIdx.x;
  const int r0  = blockIdx.x * ROWS;
  const int k2  = tid * 2;

  // Knot + reciprocal table (exact fp32 divisions, once).
  if (tid >= 2 && tid < 20) {
    int j = tid;
    float tjm1 = c_T[j-1], tj = c_T[j], tj1 = c_T[j+1], tj2 = c_T[j+2];
    jtab[j][0] = tjm1; jtab[j][1] = tj; jtab[j][2] = tj1; jtab[j][3] = tj2;
    jtab[j][4] = 1.0f / (tj1 - tj);
    jtab[j][5] = 1.0f / (tj1 - tjm1);
    jtab[j][6] = 1.0f / (tj2 - tj);
    jtab[j][7] = 0.0f;
  }

  // Async-stage input tile (ROWS x 784) into bufA; zero the padded columns.
  {
    const char* xrow  = (const char*)(X + (size_t)r0 * NIN);
    unsigned    abase = (unsigned)(size_t)bufA;
    for (int idx = tid; idx < ROWS*F4_PER_ROW; idx += NTHREADS) {
      int r = idx / F4_PER_ROW;
      int q = idx - r * F4_PER_ROW;
      async_ld_b128(abase + (unsigned)(r*FWIDTH + q*4)*4u,
                    xrow + ((size_t)r*NIN + (size_t)q*4)*4u);
    }
    for (int idx = tid; idx < ROWS*PADCOLS; idx += NTHREADS) {
      int r = idx / PADCOLS;
      int c = idx - r*PADCOLS;
      bufA[r*FWIDTH + NIN + c] = 0.0f;
    }
  }
  wait_async0();
  __syncthreads();

  float* cur = bufA;
  float* nxt = bufB;
  // even stages d=2q need cols {tid, tid+256}; odd stages d=2q+1 need {2tid, 2tid+1}
  float eA_c[2][5], eA_s[2][5], oB_c[2][5], oB_s[2][5];

  for (int layer = 0; layer < NDEPTH; ++layer) {
    // Kick off async staging of this layer's spline coefs; waited at last fused stage.
    if (layer < NDEPTH-1) {
      const char* cg    = (const char*)(sc + (size_t)layer * FHALF * SDIM);
      unsigned    cbase = (unsigned)(size_t)coefLDS;
      for (int idx = tid; idx < (FHALF*SDIM)/4; idx += NTHREADS)
        async_ld_b128(cbase + (unsigned)idx*16u, cg + (size_t)idx*16u);
    }
    // Rotation cos/sin -> registers (coalesced from precomputed table).
    if (USE_TAB) {
      #pragma unroll
      for (int q = 0; q < 5; ++q) {
        const float2* fe = (const float2*)tab + ((size_t)layer*NBDEPTH + 2*q)*FHALF;
        float2 a  = fe[tid];       eA_c[0][q]=a.x;  eA_s[0][q]=a.y;
        float2 b2 = fe[tid+256];   eA_c[1][q]=b2.x; eA_s[1][q]=b2.y;
        const float4* fo = (const float4*)tab + ((size_t)layer*NBDEPTH + 2*q+1)*(FHALF/2);
        float4 c4 = fo[tid];       oB_c[0][q]=c4.x; oB_s[0][q]=c4.y;
                                   oB_c[1][q]=c4.z; oB_s[1][q]=c4.w;
      }
    } else {
      const float* tb0 = bp + ((size_t)layer*FHALF + tid)*NBDEPTH;
      const float* tb1 = bp + ((size_t)layer*FHALF + tid+256)*NBDEPTH;
      const float* tb2 = bp + ((size_t)layer*FHALF + 2*tid)*NBDEPTH;
      const float* tb3 = tb2 + NBDEPTH;
      #pragma unroll
      for (int q = 0; q < 5; ++q) {
        sincosf(tb0[2*q],   &eA_s[0][q], &eA_c[0][q]);
        sincosf(tb1[2*q],   &eA_s[1][q], &eA_c[1][q]);
        sincosf(tb2[2*q+1], &oB_s[0][q], &oB_c[0][q]);
        sincosf(tb3[2*q+1], &oB_s[1][q], &oB_c[1][q]);
      }
    }

    // 5 fused double-stages: one LDS round trip per two butterfly stages.
    #pragma unroll
    for (int q = 0; q < 5; ++q) {
      #pragma unroll
      for (int row = 0; row < ROWS; ++row) {
        const float* cr = cur + row*FWIDTH;
        float i0 = cr[tid];        // in[m],     m = tid
        float i1 = cr[tid+256];    // in[m+256]
        float i2 = cr[tid+512];    // in[m+512]
        float i3 = cr[tid+768];    // in[m+768]
        float cA0 = eA_c[0][q], sA0 = eA_s[0][q];
        float cA1 = eA_c[1][q], sA1 = eA_s[1][q];
        float m0 = fmaf(i0, cA0,  i2*sA0);     // mid[2m]
        float m1 = fmaf(i2, cA0, -(i0*sA0));   // mid[2m+1]
        float m2 = fmaf(i1, cA1,  i3*sA1);     // mid[2m+512]
        float m3 = fmaf(i3, cA1, -(i1*sA1));   // mid[2m+513]
        float cB0 = oB_c[0][q], sB0 = oB_s[0][q];
        float cB1 = oB_c[1][q], sB1 = oB_s[1][q];
        float4 o;
        o.x = fmaf(m0, cB0,  m2*sB0);          // final[4m]
        o.y = fmaf(m2, cB0, -(m0*sB0));        // final[4m+1]
        o.z = fmaf(m1, cB1,  m3*sB1);          // final[4m+2]
        o.w = fmaf(m3, cB1, -(m1*sB1));        // final[4m+3]
        *(float4*)(nxt + row*FWIDTH + 4*tid) = o;
      }
      if (q == 4) wait_async0();               // coefs landed before spline
      __syncthreads();
      float* t = cur; cur = nxt; nxt = t;
    }

    // X0 += spline(X1) . coefs   (layers 0..6)
    if (layer < NDEPTH-1) {
      #pragma unroll
      for (int row = 0; row < ROWS; ++row) {
        float2 xv  = *(const float2*)(cur + row*FWIDTH + FHALF + k2);
        float2 x0v = *(const float2*)(cur + row*FWIDTH + k2);
        x0v.x += spline_eval(xv.x, coefLDS + (size_t)k2*SDIM,     jtab);
        x0v.y += spline_eval(xv.y, coefLDS + (size_t)(k2+1)*SDIM, jtab);
        *(float2*)(cur + row*FWIDTH + k2) = x0v;
      }
      __syncthreads();
    }
  }

  // Emit first 784 columns of each row.
  for (int idx = tid; idx < ROWS*F4_PER_ROW; idx += NTHREADS) {
    int r = idx / F4_PER_ROW;
    int q = idx - r * F4_PER_ROW;
    float4 v = *(const float4*)&cur[r*FWIDTH + q*4];
    *(float4*)(out + (size_t)(r0+r)*NOUT + (size_t)q*4) = v;
  }
}

extern "C" void kernel_launch(void* const* d_in, const int* in_sizes, int n_in,
                              void* d_out, int out_size, void* d_ws, size_t ws_size,
                              hipStream_t stream) {
  (void)in_sizes; (void)n_in; (void)out_size;
  const float* X  = (const float*)d_in[0];
  const float* bp = (const float*)d_in[1];   // [8, 512, 10]
  const float* sc = (const float*)d_in[2];   // [7, 512, 20]
  float* out = (float*)d_out;                // [4096, 784]

  if (ws_size >= CS_TAB_BYTES) {
    float2* tab = (float2*)d_ws;
    hipLaunchKernelGGL(sincos_tab_kernel,
                       dim3((NDEPTH*FHALF*NBDEPTH + NTHREADS-1)/NTHREADS),
                       dim3(NTHREADS), 0, stream, bp, tab);
    hipLaunchKernelGGL((revnet_kernel<true>), dim3(NBATCH/ROWS), dim3(NTHREADS),
                       0, stream, X, bp, (const float2*)tab, sc, out);
  } else {
    hipLaunchKernelGGL((revnet_kernel<false>), dim3(NBATCH/ROWS), dim3(NTHREADS),
                       0, stream, X, bp, (const float2*)nullptr, sc, out);
  }
}